// mutualatts_32238024524118
// MI455X (gfx1250) — compile-verified
//
#include <hip/hip_runtime.h>
#include <hip/hip_bf16.h>
#include <math.h>

typedef __attribute__((ext_vector_type(2))) float v2f;
typedef __attribute__((ext_vector_type(8))) float v8f;

#define BH  32          // batch
#define LL  2048        // sequence
#define HH  768         // hidden
#define W2S 2304        // W2 row stride (3*H)
#define PAD 772         // LDS row stride (768+4): 4 mod 64 banks -> conflict-free b64
// Dynamic LDS layout (float offsets)
#define OFF_A    0                    // A tile: 64 rows x PAD   = 49408 floats
#define OFF_B0   (64 * PAD)           // B buf0: 16 cols x PAD   = 12352 floats
#define OFF_B1   (OFF_B0 + 16 * PAD)  // B buf1: 16 cols x PAD
#define OFF_PART (OFF_B1 + 16 * PAD)  // part:   8*32*8          =  2048 floats
#define SMEM_FLOATS (OFF_PART + 8 * 32 * 8)
#define SMEM_BYTES  (SMEM_FLOATS * 4) // 304640 B < 320 KB WGP LDS

// Branch-free tanh from TRANS32 hardware ops:
//   tanh(x) = (e^{2x}-1)/(e^{2x}+1) = 1 - 2*rcp(e^{2x}+1)
// v_exp_f32 + v_rcp_f32, 5 instrs, saturates correctly at +-1, propagates NaN.
// Prefer a native tanh builtin if the toolchain ever exposes one.
#if defined(__has_builtin)
#if __has_builtin(__builtin_amdgcn_tanh_f32)
#define FAST_TANH(x) __builtin_amdgcn_tanh_f32(x)
#endif
#endif
#ifndef FAST_TANH
__device__ __forceinline__ float fast_tanh_impl(float x) {
    float e = __expf(2.0f * x);                   // v_exp_f32 (TRANS32)
    float r = __builtin_amdgcn_rcpf(e + 1.0f);    // v_rcp_f32 (TRANS32)
    return 1.0f - 2.0f * r;                       // exact algebraic identity
}
#define FAST_TANH(x) fast_tanh_impl(x)
#endif

// ---------------------------------------------------------------------------
// Kernel 1: bias_bh[b,g] = sum_h in1[b,h]*W2[g,h] + in2[b,h]*W2[g,H+h] + b2[g]
// ---------------------------------------------------------------------------
__global__ __launch_bounds__(256)
void bias_kernel(const float* __restrict__ in1, const float* __restrict__ in2,
                 const float* __restrict__ W2, const float* __restrict__ b2,
                 float* __restrict__ biasBH)
{
    int t = blockIdx.x * 256 + threadIdx.x;   // 24576 threads
    int b = t & 31;
    int g = t >> 5;
    const float* wr = W2 + (size_t)g * W2S;
    const float* x1 = in1 + b * HH;
    const float* x2 = in2 + b * HH;
    float s = b2[g];
    for (int h = 0; h < HH; ++h)
        s += x1[h] * wr[h] + x2[h] * wr[HH + h];
    biasBH[b * HH + g] = s;
}

// ---------------------------------------------------------------------------
// Kernel 2 (fused WMMA, LDS-tiled): WG = 64 text rows (4 M-tiles), 8 waves.
//   logit[m] = b1 + sum_g W1[g] * tanh( sum_h text[m,h]*W2c[g,h] + bias[b,g] )
// A (64x768) is LDS-resident for the kernel; B tiles staged to LDS per round
// (two at a time, even/odd j), so W2c is read from L2 once per WG. Both WMMA
// operands come from LDS (ds_load_2addr_b64 pairs, conflict-free stride 772).
//
// Lane layouts (CDNA5 ISA 7.12.2):
//   A 16x4:  lanes 0-15 -> M=lane, K=k0+{0,1}; lanes 16-31 -> M=lane-16, K=k0+{2,3}
//   B 4x16:  lanes 0-15 -> N=lane, K=k0+{0,1}; lanes 16-31 -> N=lane-16, K=k0+{2,3}
//   C 16x16: VGPR v -> M = v + 8*(lane/16), N = lane%16
// ---------------------------------------------------------------------------
__global__ __launch_bounds__(256)
void mutual_gemm_tanh_dot(const float* __restrict__ text,
                          const float* __restrict__ W2,
                          const float* __restrict__ W1,
                          const float* __restrict__ b1,
                          const float* __restrict__ biasBH,
                          float* __restrict__ logits)
{
    extern __shared__ float smem[];
    float* ldsA   = smem + OFF_A;
    float* ldsB   = smem + OFF_B0;      // two buffers, stride 16*PAD
    float* partsm = smem + OFF_PART;

    const int tid    = threadIdx.x;
    const int wave   = tid >> 5;
    const int lane   = tid & 31;
    const int laneLo = lane & 15;
    const int laneHi = lane >> 4;
    const int mtile  = wave & 3;        // which 16-row M-tile this wave owns
    const int parity = wave >> 2;       // j parity this wave handles

    const int rowBase = blockIdx.x * 64;    // flat (b*L + l) row
    const int bIdx    = rowBase >> 11;      // row / 2048 (64 rows stay in one b)

    // Stage 64x768 A tile (49152 contiguous floats) into padded LDS.
    const float* src = text + (size_t)rowBase * HH;
    for (int i = tid; i < 64 * HH; i += 256) {
        int m = i / HH;
        int k = i - m * HH;
        ldsA[m * PAD + k] = src[i];
    }

    float p[8];
    #pragma unroll
    for (int v = 0; v < 8; ++v) p[v] = 0.f;

    const float* aBase = ldsA + (mtile * 16 + laneLo) * PAD + 2 * laneHi;
    const float* bRead = ldsB + parity * (16 * PAD) + laneLo * PAD + 2 * laneHi;

    for (int r = 0; r < 24; ++r) {
        // Cooperative stage of B tiles j=2r (buf0) and j=2r+1 (buf1):
        // 24576 floats, coalesced 768-float runs along W2 rows.
        for (int i = tid; i < 2 * 16 * HH; i += 256) {
            int tile = i / (16 * HH);
            int rem  = i - tile * (16 * HH);
            int col  = rem / HH;
            int k    = rem - col * HH;
            int n    = (2 * r + tile) * 16 + col;
            ldsB[tile * (16 * PAD) + col * PAD + k] =
                W2[(size_t)n * W2S + 2 * HH + k];
        }
        __syncthreads();

        const int j = 2 * r + parity;
        const int n = j * 16 + laneLo;

        v8f acc = {0.f, 0.f, 0.f, 0.f, 0.f, 0.f, 0.f, 0.f};
        #pragma unroll 4
        for (int k0 = 0; k0 < HH; k0 += 4) {
            v2f a  = *(const v2f*)(aBase + k0);   // ds_load (A, LDS-resident)
            v2f bf = *(const v2f*)(bRead + k0);   // ds_load (B tile, staged)
            acc = __builtin_amdgcn_wmma_f32_16x16x4_f32(
                      false, a, false, bf, (short)0, acc, false, false);
        }

        const float bsv = biasBH[bIdx * HH + n];
        const float w1v = W1[n];
        #pragma unroll
        for (int v = 0; v < 8; ++v)
            p[v] += w1v * FAST_TANH(acc[v] + bsv); // row = mtile*16 + v + 8*laneHi

        __syncthreads();                           // compute done before restage
    }

    #pragma unroll
    for (int v = 0; v < 8; ++v) partsm[(wave * 32 + lane) * 8 + v] = p[v];
    __syncthreads();

    // Deterministic fixed-order reduction: thread t (<64) owns row t.
    if (tid < 64) {
        const int mt   = tid >> 4;          // M-tile of this row
        const int half = (tid >> 3) & 1;    // rows 0-7 from lanes 0-15, 8-15 from 16-31
        const int v    = tid & 7;
        float s = b1[0];
        #pragma unroll
        for (int wp = 0; wp < 2; ++wp) {    // waves mt and mt+4 hold this M-tile
            const int w = mt + 4 * wp;
            for (int ln = half * 16; ln < half * 16 + 16; ++ln)
                s += partsm[(w * 32 + ln) * 8 + v];
        }
        logits[rowBase + tid] = s;
    }
}

// ---------------------------------------------------------------------------
// Kernel 3: masked softmax over L, deterministic tree reductions.
// ---------------------------------------------------------------------------
__global__ __launch_bounds__(256)
void softmax_kernel(const float* __restrict__ logits,
                    const float* __restrict__ textmask,
                    float* __restrict__ att)
{
    __shared__ float red[256];
    const int b = blockIdx.x;
    const int tid = threadIdx.x;

    float loc[8];
    float lmax = -3.4e38f;
    #pragma unroll
    for (int i = 0; i < 8; ++i) {
        int l = tid + i * 256;
        float x = logits[b * LL + l] + (1.f - textmask[b * LL + l]) * -1e20f;
        loc[i] = x;
        lmax = fmaxf(lmax, x);
    }
    red[tid] = lmax; __syncthreads();
    for (int s = 128; s > 0; s >>= 1) {
        if (tid < s) red[tid] = fmaxf(red[tid], red[tid + s]);
        __syncthreads();
    }
    const float m = red[0];
    __syncthreads();

    float lsum = 0.f;
    #pragma unroll
    for (int i = 0; i < 8; ++i) { loc[i] = __expf(loc[i] - m); lsum += loc[i]; }
    red[tid] = lsum; __syncthreads();
    for (int s = 128; s > 0; s >>= 1) {
        if (tid < s) red[tid] += red[tid + s];
        __syncthreads();
    }
    const float inv = 1.f / red[0];
    #pragma unroll
    for (int i = 0; i < 8; ++i)
        att[b * LL + tid + i * 256] = loc[i] * inv;
}

// ---------------------------------------------------------------------------
// Kernel 4: partial context over L-chunks of 128 (coalesced text pass #2).
// ---------------------------------------------------------------------------
__global__ __launch_bounds__(256)
void ctx_partial_kernel(const float* __restrict__ text,
                        const float* __restrict__ att,
                        float* __restrict__ partial)
{
    const int b   = blockIdx.x >> 4;
    const int c   = blockIdx.x & 15;
    const int tid = threadIdx.x;
    const int l0  = c * 128;

    float a0 = 0.f, a1 = 0.f, a2 = 0.f;
    const float* arow = att + b * LL + l0;
    const float* trow = text + (size_t)(b * LL + l0) * HH;
    for (int l = 0; l < 128; ++l) {
        const float a = arow[l];
        a0 += a * trow[tid];
        a1 += a * trow[tid + 256];
        a2 += a * trow[tid + 512];
        trow += HH;
    }
    float* prow = partial + (size_t)blockIdx.x * HH;
    prow[tid] = a0; prow[tid + 256] = a1; prow[tid + 512] = a2;
}

// ---------------------------------------------------------------------------
// Kernel 5: deterministic fixed-order reduce of the 16 L-chunks.
// ---------------------------------------------------------------------------
__global__ __launch_bounds__(256)
void ctx_reduce_kernel(const float* __restrict__ partial,
                       float* __restrict__ ctx)
{
    const int t = blockIdx.x * 256 + threadIdx.x;  // 24576
    const int b = t / HH;
    const int h = t - b * HH;
    float s = 0.f;
    for (int c = 0; c < 16; ++c)
        s += partial[(size_t)(b * 16 + c) * HH + h];
    ctx[t] = s;
}

// ---------------------------------------------------------------------------
extern "C" void kernel_launch(void* const* d_in, const int* in_sizes, int n_in,
                              void* d_out, int out_size, void* d_ws, size_t ws_size,
                              hipStream_t stream) {
    const float* in1  = (const float*)d_in[0];
    const float* in2  = (const float*)d_in[1];
    const float* text = (const float*)d_in[2];
    const float* mask = (const float*)d_in[3];
    const float* W2   = (const float*)d_in[4];
    const float* b2   = (const float*)d_in[5];
    const float* W1   = (const float*)d_in[6];
    const float* b1   = (const float*)d_in[7];

    float* ctx = (float*)d_out;                // context: 32*768 = 24576 floats
    float* att = (float*)d_out + BH * HH;      // att:     32*2048 = 65536 floats

    float* ws      = (float*)d_ws;
    float* logits  = ws;                       // 65536 floats
    float* partial = ws + BH * LL;             // 512*768 = 393216 floats
    float* biasBH  = partial + 512 * HH;       // 24576 floats  (~1.93 MB total)

    bias_kernel         <<<  96, 256, 0, stream>>>(in1, in2, W2, b2, biasBH);
    mutual_gemm_tanh_dot<<<1024, 256, SMEM_BYTES, stream>>>(text, W2, W1, b1,
                                                            biasBH, logits);
    softmax_kernel      <<<  32, 256, 0, stream>>>(logits, mask, att);
    ctx_partial_kernel  <<< 512, 256, 0, stream>>>(text, att, partial);
    ctx_reduce_kernel   <<<  96, 256, 0, stream>>>(partial, ctx);
}